// Traj2Stroke_26199300505624
// MI455X (gfx1250) — compile-verified
//
#include <hip/hip_runtime.h>
#include <math.h>

typedef __attribute__((ext_vector_type(2))) float v2f;
typedef __attribute__((ext_vector_type(8))) float v8f;

#define CANVAS 512
#define NSEG   127      // 128 points -> 127 segments
#define BIGF   1e30f
#define EPS_F  1e-5f

// One wave32 handles a tile of 16 pixels (one row, 16 consecutive x).
// Loops over 8 blocks of 16 segments; per block: 2x V_WMMA_F32_16X16X4_F32
// (augmented-coordinate rank-2 GEMM) + VALU epilogue (clamp t, dist^2, min).
__global__ __launch_bounds__(256) void traj2stroke_wmma(
    const float* __restrict__ traj,       // 128 x 2, in [0,1)
    const float* __restrict__ thickness,  // 1
    const float* __restrict__ dropoff,    // 1
    float* __restrict__ out)              // 512 x 512
{
    const int lane      = threadIdx.x & 31;
    const int waveInBlk = threadIdx.x >> 5;
    const int gw        = blockIdx.x * 8 + waveInBlk;  // 16384 waves total
    const int y         = gw >> 5;                     // pixel row
    const int x0        = (gw & 31) << 4;              // tile start column

    const bool hi = (lane >= 16);
    const int  nl = lane & 15;

    // A matrix (16x4, f32): row M = [px, py, 1, 0].
    // Wave32 layout: lanes 0-15 hold (K=0,K=1), lanes 16-31 hold (K=2,K=3).
    v2f a;
    a.x = hi ? 1.0f : (float)(x0 + nl);
    a.y = hi ? 0.0f : (float)y;

    float m[8];
#pragma unroll
    for (int j = 0; j < 8; ++j) m[j] = BIGF;

#pragma unroll
    for (int blk = 0; blk < 8; ++blk) {
        const int  s     = blk * 16 + nl;    // this lane's segment / column N
        const bool valid = (s < NSEG);
        const int  sl    = valid ? s : 0;

        const float vx = traj[2 * sl + 0] * (float)CANVAS;
        const float vy = traj[2 * sl + 1] * (float)CANVAS;
        const float wx = traj[2 * sl + 2] * (float)CANVAS;
        const float wy = traj[2 * sl + 3] * (float)CANVAS;
        const float dx = valid ? (wx - vx) : 0.0f;
        const float dy = valid ? (wy - vy) : 0.0f;
        const float dd    = dx * dx + dy * dy;                  // ||d||^2
        const float invd2 = __builtin_amdgcn_rcpf(dd + EPS_F);  // v_rcp_f32

        // B1 column N: [dx, dy, -(v.d), 0]^T  -> C1 = (p-v).d = t_num
        // B2 column N: [-2vx, -2vy, ||v||^2, 0]^T -> C2 = ||p-v||^2 - ||p||^2
        // B layout mirrors A: lanes 0-15 rows (K=0,K=1), lanes 16-31 rows (K=2,K=3).
        v2f b1, b2;
        b1.x = hi ? (valid ? -(vx * dx + vy * dy) : 0.0f) : dx;
        b1.y = hi ? 0.0f : dy;
        b2.x = hi ? (valid ? (vx * vx + vy * vy) : BIGF) : (valid ? -2.0f * vx : 0.0f);
        b2.y = hi ? 0.0f : (valid ? -2.0f * vy : 0.0f);

        v8f cz = {};
        v8f c1 = __builtin_amdgcn_wmma_f32_16x16x4_f32(
            false, a, false, b1, (short)0, cz, false, false);
        v8f c2 = __builtin_amdgcn_wmma_f32_16x16x4_f32(
            false, a, false, b2, (short)0, cz, false, false);

        // Epilogue: lane's C/D column N == nl == this lane's segment.
#pragma unroll
        for (int j = 0; j < 8; ++j) {
            const float tn = c1[j];
            float t = tn * invd2;
            t = fminf(fmaxf(t, 0.0f), 1.0f);
            const float val = c2[j] + t * (t * dd - 2.0f * tn);
            m[j] = fminf(m[j], val);
        }
    }

    // Scalar shaping parameters (10 * sigmoid).
    const float th = 10.0f / (1.0f + __expf(-thickness[0]));
    const float de = 10.0f / (1.0f + __expf(-dropoff[0]));

#pragma unroll
    for (int j = 0; j < 8; ++j) {
        // Min across the 16 columns held within each 16-lane half.
        float v = m[j];
        v = fminf(v, __shfl_xor(v, 1, 32));
        v = fminf(v, __shfl_xor(v, 2, 32));
        v = fminf(v, __shfl_xor(v, 4, 32));
        v = fminf(v, __shfl_xor(v, 8, 32));

        const int   M     = j + (hi ? 8 : 0);   // pixel index within tile
        const float px    = (float)(x0 + M);
        const float py    = (float)y;
        const float dist2 = v + px * px + py * py;   // add back ||p||^2
        const float dist  = __builtin_amdgcn_sqrtf(fmaxf(dist2, 0.0f));

        float dk = fminf(fmaxf((th - dist) / th, 0.0f), 1.0f);
        // dk^de = exp2(de * log2(dk)); dk in [0,1], de > 0; log2(0) -> -inf -> 0.
        dk = __builtin_amdgcn_exp2f(de * __builtin_amdgcn_logf(dk));
        const float res = 1.0f - dk;

        if (nl == 0) out[y * CANVAS + x0 + M] = res;  // lane 0 & lane 16 write
    }
}

extern "C" void kernel_launch(void* const* d_in, const int* in_sizes, int n_in,
                              void* d_out, int out_size, void* d_ws, size_t ws_size,
                              hipStream_t stream) {
    const float* traj = (const float*)d_in[0];
    const float* th   = (const float*)d_in[1];
    const float* dr   = (const float*)d_in[2];
    float* out        = (float*)d_out;

    // 512 rows x 32 tiles = 16384 waves; 8 waves per 256-thread block.
    dim3 grid(2048), block(256);
    hipLaunchKernelGGL(traj2stroke_wmma, grid, block, 0, stream,
                       traj, th, dr, out);
}